// GraphSage_29798483100072
// MI455X (gfx1250) — compile-verified
//
#include <hip/hip_runtime.h>
#include <hip/hip_bf16.h>

#define N_NODES 50000
#define N_EDGES 800000
#define DIM     64
#define ROW_TILES (N_NODES / 16)   // 3125, exact

typedef __attribute__((ext_vector_type(2))) float v2f;
typedef __attribute__((ext_vector_type(8))) float v8f;

// ---------------- zero scratch ----------------
__global__ __launch_bounds__(256) void zero_kernel(float* __restrict__ p, int n) {
    int i = blockIdx.x * blockDim.x + threadIdx.x;
    if (i < n) p[i] = 0.0f;
}

// ---------------- edge scatter: agg[dst] += h[src]; cnt[dst] += 1 ----------------
// 16 threads per edge, each handles 4 consecutive dims (float4 gather, 4 fp32 atomics).
__global__ __launch_bounds__(256) void scatter_kernel(const float* __restrict__ h,
                                                      const int* __restrict__ ei,
                                                      float* __restrict__ agg,
                                                      float* __restrict__ cnt) {
    long long gid = (long long)blockIdx.x * blockDim.x + threadIdx.x;
    if (gid >= (long long)N_EDGES * 16) return;
    int e = (int)(gid >> 4);
    int t = (int)(gid & 15);
    int src = ei[e];
    int dst = ei[N_EDGES + e];
    const float4 v = *(const float4*)(h + (long long)src * DIM + t * 4);
    float* a = agg + (long long)dst * DIM + t * 4;
    unsafeAtomicAdd(a + 0, v.x);
    unsafeAtomicAdd(a + 1, v.y);
    unsafeAtomicAdd(a + 2, v.z);
    unsafeAtomicAdd(a + 3, v.w);
    if (t == 0) unsafeAtomicAdd(cnt + dst, 1.0f);
}

// ---------------- fused SAGE GEMM via V_WMMA_F32_16X16X4_F32 ----------------
// out[16x64 tile] = (agg/max(cnt,1)) @ Wl^T + bl + h @ Wr^T  (+ optional ReLU)
// One wave per 16-row tile; 4 v8f accumulators cover all 64 output columns.
__global__ __launch_bounds__(256) void sage_gemm_kernel(const float* __restrict__ h,
                                                        const float* __restrict__ agg,
                                                        const float* __restrict__ cnt,
                                                        const float* __restrict__ Wl,
                                                        const float* __restrict__ bl,
                                                        const float* __restrict__ Wr,
                                                        float* __restrict__ out,
                                                        int relu) {
    const int lane = threadIdx.x & 31;
    const int wave = threadIdx.x >> 5;
    const int tile = blockIdx.x * 8 + wave;      // wave-uniform
    if (tile >= ROW_TILES) return;               // whole-wave exit; EXEC stays all-1s in WMMA

    const int m    = lane & 15;                  // A row within tile / B column within n-tile
    const int half = lane >> 4;                  // K-half select (ISA 32-bit A/B frag layout)
    const int row  = tile * 16 + m;

    const float c   = cnt[row];
    const float inv = 1.0f / fmaxf(c, 1.0f);

    v8f acc[4] = {};                             // zero accumulators

    // --- K part 1: normalized aggregate x Wl^T  (B[k][n] = Wl[n*64 + k]) ---
    const float* Arow = agg + (long long)row * DIM;
    #pragma unroll
    for (int kk = 0; kk < 16; ++kk) {
        const int k = 4 * kk + 2 * half;
        v2f a;
        a.x = Arow[k] * inv;
        a.y = Arow[k + 1] * inv;
        #pragma unroll
        for (int t = 0; t < 4; ++t) {
            const int n = t * 16 + m;
            v2f b;
            b.x = Wl[n * DIM + k];
            b.y = Wl[n * DIM + k + 1];
            acc[t] = __builtin_amdgcn_wmma_f32_16x16x4_f32(
                false, a, false, b, (short)0, acc[t], false, false);
        }
    }

    // --- K part 2: h x Wr^T ---
    const float* Hrow = h + (long long)row * DIM;
    #pragma unroll
    for (int kk = 0; kk < 16; ++kk) {
        const int k = 4 * kk + 2 * half;
        v2f a;
        a.x = Hrow[k];
        a.y = Hrow[k + 1];
        #pragma unroll
        for (int t = 0; t < 4; ++t) {
            const int n = t * 16 + m;
            v2f b;
            b.x = Wr[n * DIM + k];
            b.y = Wr[n * DIM + k + 1];
            acc[t] = __builtin_amdgcn_wmma_f32_16x16x4_f32(
                false, a, false, b, (short)0, acc[t], false, false);
        }
    }

    // --- epilogue: bias + optional ReLU; D layout: row = j + 8*half, col = t*16 + (lane&15) ---
    #pragma unroll
    for (int t = 0; t < 4; ++t) {
        const int   col  = t * 16 + m;
        const float bias = bl[col];
        #pragma unroll
        for (int j = 0; j < 8; ++j) {
            float v = acc[t][j] + bias;
            if (relu) v = fmaxf(v, 0.0f);
            out[(long long)(tile * 16 + 8 * half + j) * DIM + col] = v;
        }
    }
}

extern "C" void kernel_launch(void* const* d_in, const int* in_sizes, int n_in,
                              void* d_out, int out_size, void* d_ws, size_t ws_size,
                              hipStream_t stream) {
    const float* x  = (const float*)d_in[0];
    const int*   ei = (const int*)d_in[1];
    const float* Wl[3] = { (const float*)d_in[2], (const float*)d_in[5], (const float*)d_in[8] };
    const float* blv[3] = { (const float*)d_in[3], (const float*)d_in[6], (const float*)d_in[9] };
    const float* Wr[3] = { (const float*)d_in[4], (const float*)d_in[7], (const float*)d_in[10] };

    // workspace layout (floats): agg[N*D] | cnt[N] | hA[N*D] | hB[N*D]
    float* agg = (float*)d_ws;
    float* cnt = agg + (size_t)N_NODES * DIM;
    float* hA  = cnt + N_NODES;
    float* hB  = hA + (size_t)N_NODES * DIM;

    const float* lay_in[3]  = { x,  hA, hB };
    float*       lay_out[3] = { hA, hB, (float*)d_out };

    const int nZero      = N_NODES * DIM + N_NODES;     // agg + cnt contiguous
    const int zeroBlocks = (nZero + 255) / 256;
    const int scatBlocks = (int)(((long long)N_EDGES * 16 + 255) / 256);
    const int gemmBlocks = (ROW_TILES + 7) / 8;

    for (int l = 0; l < 3; ++l) {
        zero_kernel<<<zeroBlocks, 256, 0, stream>>>(agg, nZero);
        scatter_kernel<<<scatBlocks, 256, 0, stream>>>(lay_in[l], ei, agg, cnt);
        sage_gemm_kernel<<<gemmBlocks, 256, 0, stream>>>(
            lay_in[l], agg, cnt, Wl[l], blv[l], Wr[l], lay_out[l], (l < 2) ? 1 : 0);
    }
}